// CausalSelfAttention_4054449128214
// MI455X (gfx1250) — compile-verified
//
#include <hip/hip_runtime.h>
#include <math.h>

typedef float v2f __attribute__((ext_vector_type(2)));
typedef float v8f __attribute__((ext_vector_type(8)));

#define TT   1024
#define CC   768
#define NH   12
#define HS   64
#define QKV_LD (3 * CC)

// ---------------------------------------------------------------------------
// fp32 WMMA GEMM: C[M,N] = A[M,K] @ B[K,N] + bias[N]
// One wave32 per 16x64 output tile (4 accumulators). Loads are issued into
// distinct registers BEFORE the 4 WMMAs so s_wait_loadcnt staggers instead of
// draining to 0 in front of every matrix op.
// ---------------------------------------------------------------------------
__global__ __launch_bounds__(256)
void gemm_wmma_f32(const float* __restrict__ A, const float* __restrict__ B,
                   const float* __restrict__ bias, float* __restrict__ C,
                   int M, int K, int N) {
    const int wave = threadIdx.x >> 5;
    const int lane = threadIdx.x & 31;
    const int half = lane >> 4;
    const int lm   = lane & 15;

    const int ntiles_n = N >> 6;                 // 64-wide column tiles
    const int total    = (M >> 4) * ntiles_n;
    const int tile     = blockIdx.x * 8 + wave;
    if (tile >= total) return;                   // uniform per wave

    const int row0 = (tile / ntiles_n) << 4;
    const int col0 = (tile % ntiles_n) << 6;

    // A fragment: lane&15 = M row; K half selected by lane>>4 (contiguous float2)
    const float* Arow = A + (size_t)(row0 + lm) * K;
    // B fragment: lane&15 = N col within a 16-wide chunk; rows k,k+1 (stride N)
    const float* Bcol = B + col0 + lm;

    v8f acc[4] = {};
    for (int k = 0; k < K; k += 4) {
        const int ka = k + 2 * half;

        // ---- load phase: distinct registers for every fragment ----------
        v2f a;
        a.x = Arow[ka];
        a.y = Arow[ka + 1];
        const float* b0 = Bcol + (size_t)ka * N;
        const float* b1 = b0 + N;
        v2f b[4];
#pragma unroll
        for (int c = 0; c < 4; ++c) {
            b[c].x = b0[c * 16];
            b[c].y = b1[c * 16];
        }
        // prefetch next K panel (lowers to global_prefetch_b8)
        __builtin_prefetch(Arow + ka + 32, 0, 1);
        __builtin_prefetch(Bcol + (size_t)(ka + 4) * N, 0, 1);

        // ---- compute phase: 4 independent accumulator chains ------------
#pragma unroll
        for (int c = 0; c < 4; ++c) {
            acc[c] = __builtin_amdgcn_wmma_f32_16x16x4_f32(
                         false, a, false, b[c], (short)0, acc[c], false, false);
        }
    }

#pragma unroll
    for (int c = 0; c < 4; ++c) {
        const float bv = bias[col0 + c * 16 + lm];
#pragma unroll
        for (int r = 0; r < 8; ++r) {
            // D layout: VGPR r -> row (r + 8*half), col = lane&15
            C[(size_t)(row0 + r + 8 * half) * N + col0 + c * 16 + lm] = acc[c][r] + bv;
        }
    }
}

// ---------------------------------------------------------------------------
// Flash-style causal attention, fp32 WMMA throughout.
// Grid: (T/16 query tiles, NH heads); one wave32 per block.
// qkv layout: [T, 3C] with q at col 0, k at col C, v at col 2C; head h at h*HS.
// y layout: [T, C].
// ---------------------------------------------------------------------------
__global__ __launch_bounds__(32)
void attn_wmma_f32(const float* __restrict__ qkv, float* __restrict__ y) {
    const int qt   = blockIdx.x;
    const int h    = blockIdx.y;
    const int lane = threadIdx.x & 31;
    const int half = lane >> 4;
    const int lm   = lane & 15;

    const float* Q  = qkv + (size_t)h * HS;
    const float* Kp = qkv + CC + (size_t)h * HS;
    const float* Vp = qkv + 2 * CC + (size_t)h * HS;

    __shared__ float Pt[16][17];   // padded to dodge bank conflicts

    // Load q tile as 16 A-fragments (K-steps of 4 over HS=64), register resident
    v2f qa[16];
    {
        const float* qrow = Q + (size_t)(qt * 16 + lm) * QKV_LD;
#pragma unroll
        for (int s = 0; s < 16; ++s) {
            const int d = s * 4 + 2 * half;
            qa[s].x = qrow[d];
            qa[s].y = qrow[d + 1];
        }
    }

    v8f acc0 = {}, acc1 = {}, acc2 = {}, acc3 = {};
    float m[8], l[8];
#pragma unroll
    for (int r = 0; r < 8; ++r) { m[r] = -INFINITY; l[r] = 0.0f; }

    const float scale = 0.125f;    // 1/sqrt(64)

    for (int kt = 0; kt <= qt; ++kt) {
        // ---- scores S = q @ k^T over this 16-key tile -----------------
        v8f S = {};
        const float* krow = Kp + (size_t)(kt * 16 + lm) * QKV_LD;
#pragma unroll
        for (int s = 0; s < 16; ++s) {
            const int d = s * 4 + 2 * half;
            v2f b;
            b.x = krow[d];
            b.y = krow[d + 1];
            S = __builtin_amdgcn_wmma_f32_16x16x4_f32(
                    false, qa[s], false, b, (short)0, S, false, false);
        }

        // ---- scale + causal mask (D layout: row = r+8*half, col = lm) --
        float p[8];
#pragma unroll
        for (int r = 0; r < 8; ++r) {
            const int qpos = qt * 16 + r + 8 * half;
            const int kpos = kt * 16 + lm;
            p[r] = (kpos <= qpos) ? (S[r] * scale) : -INFINITY;
        }

        // ---- online softmax: rowmax / correction / rowsum --------------
        float corr[8];
#pragma unroll
        for (int r = 0; r < 8; ++r) {
            float v = p[r];
            v = fmaxf(v, __shfl_xor(v, 1, 32));
            v = fmaxf(v, __shfl_xor(v, 2, 32));
            v = fmaxf(v, __shfl_xor(v, 4, 32));
            v = fmaxf(v, __shfl_xor(v, 8, 32));
            const float mn = fmaxf(m[r], v);
            corr[r] = __expf(m[r] - mn);
            const float e = __expf(p[r] - mn);    // masked -> exp(-inf)=0
            float s = e;
            s += __shfl_xor(s, 1, 32);
            s += __shfl_xor(s, 2, 32);
            s += __shfl_xor(s, 4, 32);
            s += __shfl_xor(s, 8, 32);
            l[r] = l[r] * corr[r] + s;
            m[r] = mn;
            Pt[r + 8 * half][lm] = e;             // D layout -> LDS
        }
#pragma unroll
        for (int r = 0; r < 8; ++r) {
            acc0[r] *= corr[r];
            acc1[r] *= corr[r];
            acc2[r] *= corr[r];
            acc3[r] *= corr[r];
        }
        __syncthreads();

        // ---- P @ V: A from LDS (A layout), 4 HS chunks x 4 K-steps ----
#pragma unroll
        for (int s = 0; s < 4; ++s) {
            const int kk = s * 4 + 2 * half;
            v2f a;
            a.x = Pt[lm][kk];
            a.y = Pt[lm][kk + 1];
            const float* v0 = Vp + (size_t)(kt * 16 + kk) * QKV_LD;
            const float* v1 = v0 + QKV_LD;
            v2f b0, b1, b2, b3;
            b0.x = v0[lm];       b0.y = v1[lm];
            b1.x = v0[16 + lm];  b1.y = v1[16 + lm];
            b2.x = v0[32 + lm];  b2.y = v1[32 + lm];
            b3.x = v0[48 + lm];  b3.y = v1[48 + lm];
            acc0 = __builtin_amdgcn_wmma_f32_16x16x4_f32(false, a, false, b0, (short)0, acc0, false, false);
            acc1 = __builtin_amdgcn_wmma_f32_16x16x4_f32(false, a, false, b1, (short)0, acc1, false, false);
            acc2 = __builtin_amdgcn_wmma_f32_16x16x4_f32(false, a, false, b2, (short)0, acc2, false, false);
            acc3 = __builtin_amdgcn_wmma_f32_16x16x4_f32(false, a, false, b3, (short)0, acc3, false, false);
        }
        __syncthreads();   // Pt reused next key-tile
    }

    // ---- epilogue: normalize by row sums, write y[T, C] ---------------
#pragma unroll
    for (int r = 0; r < 8; ++r) {
        const float inv = 1.0f / l[r];
        const size_t base = (size_t)(qt * 16 + r + 8 * half) * CC + (size_t)h * HS;
        y[base +  0 + lm] = acc0[r] * inv;
        y[base + 16 + lm] = acc1[r] * inv;
        y[base + 32 + lm] = acc2[r] * inv;
        y[base + 48 + lm] = acc3[r] * inv;
    }
}

// ---------------------------------------------------------------------------
extern "C" void kernel_launch(void* const* d_in, const int* in_sizes, int n_in,
                              void* d_out, int out_size, void* d_ws, size_t ws_size,
                              hipStream_t stream) {
    (void)in_sizes; (void)n_in; (void)out_size; (void)ws_size;
    const float* x      = (const float*)d_in[0];   // [1,1024,768]
    const float* W_attn = (const float*)d_in[1];   // [768, 2304]
    const float* b_attn = (const float*)d_in[2];   // [2304]
    const float* W_proj = (const float*)d_in[3];   // [768, 768]
    const float* b_proj = (const float*)d_in[4];   // [768]
    float* out = (float*)d_out;                    // [1,1024,768]

    float* qkv = (float*)d_ws;                       // [1024, 2304]
    float* y   = qkv + (size_t)TT * QKV_LD;          // [1024, 768]

    {   // qkv = x @ W_attn + b_attn   (16x64 tiles: 64 * 36 = 2304 tiles)
        const int tiles = (TT / 16) * (QKV_LD / 64);
        gemm_wmma_f32<<<(tiles + 7) / 8, 256, 0, stream>>>(
            x, W_attn, b_attn, qkv, TT, CC, QKV_LD);
    }
    {   // per-head causal attention (forward value of the straight-through trick)
        attn_wmma_f32<<<dim3(TT / 16, NH), 32, 0, stream>>>(qkv, y);
    }
    {   // out = y @ W_proj + b_proj   (16x64 tiles: 64 * 12 = 768 tiles)
        const int tiles = (TT / 16) * (CC / 64);
        gemm_wmma_f32<<<(tiles + 7) / 8, 256, 0, stream>>>(
            y, W_proj, b_proj, out, TT, CC, CC);
    }
}